// crossgatconv_70600672411867
// MI455X (gfx1250) — compile-verified
//
#include <hip/hip_runtime.h>
#include <hip/hip_bf16.h>
#include <math.h>

#define N_NODES 50000
#define N_EDGES 800000
#define CH 64
#define NF (N_NODES * CH)

typedef float v2f __attribute__((ext_vector_type(2)));
typedef float v8f __attribute__((ext_vector_type(8)));

__device__ __forceinline__ float sigmoidf(float v) {
    return 1.0f / (1.0f + __expf(-v));
}

// ---------------------------------------------------------------------------
// Kernel A: per-node minmax norm (in LDS) + three 64x64 projections via
// V_WMMA_F32_16X16X4_F32. One block = 16 nodes, 4 waves; wave w computes
// output columns [16w, 16w+16) for W1, W2, Wres. Also zero-inits the atomic
// accumulator arrays for its 16 nodes (graph-replay safe re-init each call).
// ---------------------------------------------------------------------------
__global__ __launch_bounds__(128)
void node_proj_kernel(const float* __restrict__ ndata,
                      const float* __restrict__ W1, const float* __restrict__ b1,
                      const float* __restrict__ W2, const float* __restrict__ b2,
                      const float* __restrict__ Wres, const float* __restrict__ bres,
                      float* __restrict__ lout, float* __restrict__ rout,
                      float* __restrict__ resout,
                      float* __restrict__ suml, float* __restrict__ sumr,
                      float* __restrict__ accr, float* __restrict__ accl)
{
    __shared__ float xs[16][65];    // stride 65 -> conflict-free column reads
    __shared__ float s_mn[16], s_sc[16];
    const int tid  = threadIdx.x;
    const int base = blockIdx.x * 16;

    // coalesced load of 16x64 node tile
    for (int i = tid; i < 16 * CH; i += 128) {
        const int row = i >> 6, col = i & 63;
        xs[row][col] = ndata[(base + row) * CH + col];
    }
    // zero the atomic accumulators for these nodes
    for (int i = tid; i < 16 * CH; i += 128) {
        const int g = base * CH + i;
        suml[g] = 0.f; sumr[g] = 0.f; accr[g] = 0.f; accl[g] = 0.f;
    }
    __syncthreads();

    // per-node min/max + scale
    if (tid < 16) {
        float mn = xs[tid][0], mx = xs[tid][0];
        for (int c = 1; c < CH; ++c) {
            const float v = xs[tid][c];
            mn = fminf(mn, v); mx = fmaxf(mx, v);
        }
        s_mn[tid] = mn;
        s_sc[tid] = 1.0f / (mx - mn + 1e-5f);
    }
    __syncthreads();

    // normalize tile in place
    for (int i = tid; i < 16 * CH; i += 128) {
        const int row = i >> 6, col = i & 63;
        xs[row][col] = (xs[row][col] - s_mn[row]) * s_sc[row];
    }
    __syncthreads();

    const int wave = tid >> 5;      // 0..3 -> 16-column output tile
    const int lane = tid & 31;
    const int half = lane >> 4;     // 0/1
    const int m16  = lane & 15;     // row (A) and column (B/C) index
    const int wcol = wave * 16;

    const float* Ws[3]   = { W1, W2, Wres };
    const float* bsv[3]  = { b1, b2, bres };
    float*       outs[3] = { lout, rout, resout };

    for (int mat = 0; mat < 3; ++mat) {
        const float* W = Ws[mat];
        v8f acc = {};
        #pragma unroll
        for (int kk = 0; kk < 16; ++kk) {
            const int kb = kk * 4 + half * 2;   // K index for vgpr 0 of A/B frag
            v2f a, b;
            // A 16x4 f32 frag: lane holds rows M=lane%16, K = kb, kb+1
            a.x = xs[m16][kb];
            a.y = xs[m16][kb + 1];
            // B 4x16 f32 frag: lane holds col N=lane%16, K = kb, kb+1
            b.x = W[(kb    ) * CH + wcol + m16];
            b.y = W[(kb + 1) * CH + wcol + m16];
            acc = __builtin_amdgcn_wmma_f32_16x16x4_f32(
                false, a, false, b, (short)0, acc, false, false);
        }
        const float bv  = bsv[mat][wcol + m16];
        float*      out = outs[mat];
        #pragma unroll
        for (int v = 0; v < 8; ++v) {
            const int row = v + half * 8;       // C/D layout: M = v (+8 for hi half)
            out[(base + row) * CH + wcol + m16] = acc[v] + bv;
        }
    }
}

// ---------------------------------------------------------------------------
// Edge pass 1: one wave per edge (incl. self-loops). Computes softmax
// numerators exp(sigmoid(l[s]+l[d])), exp(sigmoid(r[s]+r[d])), accumulates
// denominators suml/sumr and first-hop numerator accr += exl * r[s].
// Max-subtraction is skipped: sigmoid output in (0,1) -> exp is safe.
// ---------------------------------------------------------------------------
__global__ __launch_bounds__(256)
void edge_pass1(const int* __restrict__ src, const int* __restrict__ dst,
                const float* __restrict__ l, const float* __restrict__ r,
                float* __restrict__ suml, float* __restrict__ sumr,
                float* __restrict__ accr)
{
    const int e = blockIdx.x * 8 + (threadIdx.x >> 5);
    if (e >= N_EDGES + N_NODES) return;
    int s, d;
    if (e < N_EDGES) { s = src[e]; d = dst[e]; }
    else             { s = e - N_EDGES; d = s; }        // self loop

    const int lane = threadIdx.x & 31;
    const float2* l2 = (const float2*)l;
    const float2* r2 = (const float2*)r;
    const float2 ls = l2[s * 32 + lane], ld = l2[d * 32 + lane];
    const float2 rs = r2[s * 32 + lane], rd = r2[d * 32 + lane];

    const float exl0 = __expf(sigmoidf(ls.x + ld.x));
    const float exl1 = __expf(sigmoidf(ls.y + ld.y));
    const float exr0 = __expf(sigmoidf(rs.x + rd.x));
    const float exr1 = __expf(sigmoidf(rs.y + rd.y));

    const int gb = d * CH + lane * 2;
    atomicAdd(&suml[gb],     exl0);        atomicAdd(&suml[gb + 1], exl1);
    atomicAdd(&sumr[gb],     exr0);        atomicAdd(&sumr[gb + 1], exr1);
    atomicAdd(&accr[gb],     exl0 * rs.x); atomicAdd(&accr[gb + 1], exl1 * rs.y);
}

// ft_r = accr / suml  (in place; suml > 0 guaranteed by self-loops)
__global__ __launch_bounds__(256)
void ftr_norm(float* __restrict__ accr, const float* __restrict__ suml)
{
    const int i = blockIdx.x * 256 + threadIdx.x;
    if (i < NF) accr[i] = accr[i] / suml[i];
}

// ---------------------------------------------------------------------------
// Edge pass 2: recompute exr = exp(sigmoid(r[s]+r[d])) (cheaper than storing
// 218 MB of per-edge weights), accumulate accl += exr * ft_r[s].
// ---------------------------------------------------------------------------
__global__ __launch_bounds__(256)
void edge_pass2(const int* __restrict__ src, const int* __restrict__ dst,
                const float* __restrict__ r, const float* __restrict__ ftr,
                float* __restrict__ accl)
{
    const int e = blockIdx.x * 8 + (threadIdx.x >> 5);
    if (e >= N_EDGES + N_NODES) return;
    int s, d;
    if (e < N_EDGES) { s = src[e]; d = dst[e]; }
    else             { s = e - N_EDGES; d = s; }

    const int lane = threadIdx.x & 31;
    const float2* r2 = (const float2*)r;
    const float2* f2 = (const float2*)ftr;
    const float2 rs = r2[s * 32 + lane], rd = r2[d * 32 + lane];
    const float2 fs = f2[s * 32 + lane];

    const float exr0 = __expf(sigmoidf(rs.x + rd.x));
    const float exr1 = __expf(sigmoidf(rs.y + rd.y));

    const int gb = d * CH + lane * 2;
    atomicAdd(&accl[gb],     exr0 * fs.x);
    atomicAdd(&accl[gb + 1], exr1 * fs.y);
}

// out = m0[ch] * accl/sumr + res
__global__ __launch_bounds__(256)
void finalize(const float* __restrict__ accl, const float* __restrict__ sumr,
              const float* __restrict__ res, const float* __restrict__ m0,
              float* __restrict__ out)
{
    const int i = blockIdx.x * 256 + threadIdx.x;
    if (i < NF) out[i] = m0[i & 63] * (accl[i] / sumr[i]) + res[i];
}

extern "C" void kernel_launch(void* const* d_in, const int* in_sizes, int n_in,
                              void* d_out, int out_size, void* d_ws, size_t ws_size,
                              hipStream_t stream) {
    const float* ndata = (const float*)d_in[0];
    const int*   src   = (const int*)  d_in[1];
    const int*   dst   = (const int*)  d_in[2];
    const float* W1    = (const float*)d_in[3];
    const float* b1    = (const float*)d_in[4];
    const float* W2    = (const float*)d_in[5];
    const float* b2    = (const float*)d_in[6];
    const float* Wres  = (const float*)d_in[7];
    const float* bres  = (const float*)d_in[8];
    const float* m0    = (const float*)d_in[9];   // m[0] = first 64 floats
    float* out = (float*)d_out;

    float* ws   = (float*)d_ws;
    float* l    = ws + (size_t)0 * NF;
    float* r    = ws + (size_t)1 * NF;
    float* res  = ws + (size_t)2 * NF;
    float* suml = ws + (size_t)3 * NF;
    float* sumr = ws + (size_t)4 * NF;
    float* accr = ws + (size_t)5 * NF;   // becomes ft_r after ftr_norm
    float* accl = ws + (size_t)6 * NF;

    // nodes: norm + projections + accumulator zeroing
    node_proj_kernel<<<N_NODES / 16, 128, 0, stream>>>(
        ndata, W1, b1, W2, b2, Wres, bres,
        l, r, res, suml, sumr, accr, accl);

    const int eblocks = (N_EDGES + N_NODES + 7) / 8;   // 8 waves (edges) / block
    edge_pass1<<<eblocks, 256, 0, stream>>>(src, dst, l, r, suml, sumr, accr);
    ftr_norm<<<(NF + 255) / 256, 256, 0, stream>>>(accr, suml);
    edge_pass2<<<eblocks, 256, 0, stream>>>(src, dst, r, accr, accl);
    finalize<<<(NF + 255) / 256, 256, 0, stream>>>(accl, sumr, res, m0, out);
}